// PRSNet_7507602833420
// MI455X (gfx1250) — compile-verified
//
#include <hip/hip_runtime.h>
#include <hip/hip_bf16.h>

typedef __attribute__((ext_vector_type(2))) float v2f;
typedef __attribute__((ext_vector_type(8))) float v8f;

#define NB 16               // batch (graphs)
#define NGENES 19836
#define DIN 11
#define DH 64
#define NNODES (NB * NGENES)        // 317376
#define NTILES (NNODES / 16)        // 19836 (exact)
#define BN_EPS 1e-5f

__device__ __forceinline__ float geluf(float v) {
    // exact erf-based GELU (torch nn.GELU default)
    return 0.5f * v * (1.0f + erff(v * 0.70710678118654752440f));
}

// ---------------- K0: init scratch, fold key/q, pre-swizzle W_gin ----------
// Bf[(nt*16+s)*32 + lane] = WMMA-B fragment of W_gin for N-tile nt, K-step s.
__global__ void k_init(const float* __restrict__ Wk, const float* __restrict__ bk,
                       const float* __restrict__ Wq, const float* __restrict__ Wg,
                       float* u, float* cbias, float* ssum, float* ssq,
                       float* S, float* wsum, v2f* __restrict__ Bf) {
    int t = threadIdx.x;  // 256 threads, 1 block
    if (t < DH) {
        float s = 0.f;
        #pragma unroll 8
        for (int n = 0; n < DH; ++n) s += Wk[t * DH + n] * Wq[n];
        u[t] = s;
        ssum[t] = 0.f;
        ssq[t]  = 0.f;
    }
    if (t == DH) {
        float s = 0.f;
        for (int n = 0; n < DH; ++n) s += bk[n] * Wq[n];
        cbias[0] = s;
    }
    for (int i = t; i < NB * DH; i += 256) S[i] = 0.f;
    if (t < NB) wsum[t] = 0.f;

    // pre-swizzled B fragments: 4 ntiles x 16 ksteps x 32 lanes (float2 each)
    for (int i = t; i < 4 * 16 * 32; i += 256) {
        int lane = i & 31;
        int s    = (i >> 5) & 15;
        int nt   = i >> 9;
        int koff = (lane >> 4) * 2;
        int col  = nt * 16 + (lane & 15);
        v2f b;
        b.x = Wg[(4 * s + koff) * DH + col];
        b.y = Wg[(4 * s + koff + 1) * DH + col];
        Bf[i] = b;
    }
}

// ---------------- K1: encoder  h = gelu(x@W_enc + b) + gene_emb ------------
__global__ void k_encoder(const float* __restrict__ x, const float* __restrict__ We,
                          const float* __restrict__ be, const float* __restrict__ gemb,
                          float* __restrict__ h, float* __restrict__ acc) {
    int gid = blockIdx.x * blockDim.x + threadIdx.x;
    if (gid >= NNODES * DH) return;
    int row = gid >> 6;
    int col = gid & 63;
    const float* xr = x + (size_t)row * DIN;
    float s = be[col];
    #pragma unroll
    for (int i = 0; i < DIN; ++i) s += xr[i] * We[i * DH + col];
    s = geluf(s);
    s += gemb[(size_t)(row % NGENES) * DH + col];
    h[gid]   = s;
    acc[gid] = s;   // acc starts as h (GIN eps=0 self term)
}

// ---------------- K2: edge scatter  acc[dst] += h[src] ---------------------
__global__ void k_scatter(const float* __restrict__ h, const int* __restrict__ esrc,
                          const int* __restrict__ edst, float* acc, int total) {
    int gid = blockIdx.x * blockDim.x + threadIdx.x;
    if (gid >= total) return;
    int e   = gid >> 6;
    int col = gid & 63;
    int s = esrc[e];
    int d = edst[e];
    unsafeAtomicAdd(&acc[(size_t)d * DH + col], h[(size_t)s * DH + col]);
}

// ---------------- K3: GIN GEMM (f32 WMMA) + fused BN stats -----------------
// In-place: reads acc rows fully into registers, then overwrites same rows.
// B fragments come pre-swizzled (coalesced b64 loads, identical per wave).
__global__ void k_gin_gemm(float* acc, const v2f* __restrict__ Bf,
                           float* ssum, float* ssq) {
    int wave = (blockIdx.x * blockDim.x + threadIdx.x) >> 5;
    int lane = threadIdx.x & 31;
    if (wave >= NTILES) return;               // wave-uniform: EXEC stays all-ones
    const int m    = lane & 15;
    const int koff = (lane >> 4) * 2;         // lanes 16-31 hold upper K pair
    const int row0 = wave * 16;

    // A fragments for the whole K=64 (16 steps x float2), register resident.
    v2f a[16];
    const float* rowp = acc + (size_t)(row0 + m) * DH;
    #pragma unroll
    for (int s = 0; s < 16; ++s) {
        a[s].x = rowp[4 * s + koff];
        a[s].y = rowp[4 * s + koff + 1];
    }

    #pragma unroll
    for (int nt = 0; nt < 4; ++nt) {
        int col = nt * 16 + m;
        const v2f* bp = Bf + nt * 16 * 32 + lane;
        v8f c = {};
        #pragma unroll
        for (int s = 0; s < 16; ++s) {
            v2f b = bp[s * 32];
            c = __builtin_amdgcn_wmma_f32_16x16x4_f32(
                    false, a[s], false, b, (short)0, c, false, false);
        }
        float ls = 0.f, lq = 0.f;
        #pragma unroll
        for (int i = 0; i < 8; ++i) {
            int r = row0 + i + 8 * (lane >> 4);   // C layout: VGPR i -> M=i(+8)
            float v = c[i];
            acc[(size_t)r * DH + col] = v;
            ls += v;
            lq += v * v;
        }
        unsafeAtomicAdd(&ssum[col], ls);
        unsafeAtomicAdd(&ssq[col], lq);
    }
}

// ---------------- K4: finalize BN scale/shift ------------------------------
__global__ void k_bnfix(const float* ssum, const float* ssq,
                        const float* __restrict__ gamma, const float* __restrict__ beta,
                        float* scale, float* shift) {
    int c = threadIdx.x;
    if (c < DH) {
        const float invN = 1.0f / (float)NNODES;
        float mu  = ssum[c] * invN;
        float var = ssq[c] * invN - mu * mu;
        float sc  = gamma[c] * rsqrtf(var + BN_EPS);
        scale[c] = sc;
        shift[c] = beta[c] - mu * sc;
    }
}

// ---------------- K5: fused readout --------------------------------------
// y = gelu(scale*h2+shift); w = sigmoid(y.u + c); S[g] += w*y; wsum[g] += w
__global__ void k_readout(const float* __restrict__ h2, const float* __restrict__ scale,
                          const float* __restrict__ shift, const float* __restrict__ u,
                          const float* __restrict__ cbias,
                          float* S, float* wsum, float* __restrict__ w_out) {
    __shared__ float Sl[2][DH];
    __shared__ float wsl[2];
    for (int t = threadIdx.x; t < 2 * DH + 2; t += blockDim.x) {
        if (t < 2 * DH) Sl[t >> 6][t & 63] = 0.f;
        else wsl[t - 2 * DH] = 0.f;
    }
    __syncthreads();

    int lane = threadIdx.x & 31;
    int tile = blockIdx.x * 8 + (threadIdx.x >> 5);
    int g0   = (blockIdx.x * 128) / NGENES;   // block spans <= 2 graphs

    if (tile < NTILES) {                       // wave-uniform guard
        const int m    = lane & 15;
        const int koff = (lane >> 4) * 2;
        const int row  = tile * 16 + m;
        const float* rowp = h2 + (size_t)row * DH;
        float y[32];
        float p = 0.f;
        #pragma unroll
        for (int s = 0; s < 16; ++s) {
            int k0 = 4 * s + koff;
            float y0 = geluf(scale[k0] * rowp[k0] + shift[k0]);
            float y1 = geluf(scale[k0 + 1] * rowp[k0 + 1] + shift[k0 + 1]);
            y[2 * s]     = y0;
            y[2 * s + 1] = y1;
            p += y0 * u[k0] + y1 * u[k0 + 1];
        }
        p += __shfl_xor(p, 16);               // combine the two K halves (wave32)
        float wv = 1.0f / (1.0f + expf(-(p + cbias[0])));
        if (lane < 16) w_out[row] = wv;

        int slot = row / NGENES - g0;         // 0 or 1
        float* Sp = Sl[slot];
        #pragma unroll
        for (int s = 0; s < 16; ++s) {
            atomicAdd(&Sp[4 * s + koff],     wv * y[2 * s]);
            atomicAdd(&Sp[4 * s + koff + 1], wv * y[2 * s + 1]);
        }
        if (lane < 16) atomicAdd(&wsl[slot], wv);
    }
    __syncthreads();
    for (int t = threadIdx.x; t < 2 * DH + 2; t += blockDim.x) {
        if (t < 2 * DH) {
            int slot = t >> 6, col = t & 63, g = g0 + slot;
            if (g < NB) unsafeAtomicAdd(&S[g * DH + col], Sl[slot][col]);
        } else {
            int slot = t - 2 * DH, g = g0 + slot;
            if (g < NB) unsafeAtomicAdd(&wsum[g], wsl[slot]);
        }
    }
}

// ---------------- K6: 16-row tail: val GEMM + predictor --------------------
__global__ void k_final(const float* __restrict__ S, const float* __restrict__ wsum,
                        const float* __restrict__ Wv, const float* __restrict__ bv,
                        const float* __restrict__ Wp1, const float* __restrict__ bp1,
                        const float* __restrict__ Wp2, const float* __restrict__ bp2,
                        float* __restrict__ preds) {
    __shared__ float gh[NB][DH];
    __shared__ float z1[NB][DH];
    int t = threadIdx.x;  // 64 threads
    if (t < DH) {
        for (int g = 0; g < NB; ++g) {
            float s = wsum[g] * bv[t];
            #pragma unroll 8
            for (int k = 0; k < DH; ++k) s += S[g * DH + k] * Wv[k * DH + t];
            gh[g][t] = s;
        }
    }
    __syncthreads();
    if (t < DH) {
        for (int g = 0; g < NB; ++g) {
            float s = bp1[t];
            #pragma unroll 8
            for (int k = 0; k < DH; ++k) s += gh[g][k] * Wp1[k * DH + t];
            z1[g][t] = geluf(s);
        }
    }
    __syncthreads();
    if (t < NB) {
        float s = bp2[0];
        #pragma unroll 8
        for (int k = 0; k < DH; ++k) s += z1[t][k] * Wp2[k];
        preds[t] = s;
    }
}

extern "C" void kernel_launch(void* const* d_in, const int* in_sizes, int n_in,
                              void* d_out, int out_size, void* d_ws, size_t ws_size,
                              hipStream_t stream) {
    const float* x     = (const float*)d_in[0];
    const int*   esrc  = (const int*)  d_in[1];
    const int*   edst  = (const int*)  d_in[2];
    // d_in[3] = covariates (empty)
    const float* W_enc = (const float*)d_in[4];
    const float* b_enc = (const float*)d_in[5];
    const float* gemb  = (const float*)d_in[6];
    const float* W_gin = (const float*)d_in[7];
    const float* gamma = (const float*)d_in[8];
    const float* beta  = (const float*)d_in[9];
    const float* W_key = (const float*)d_in[10];
    const float* b_key = (const float*)d_in[11];
    const float* W_q   = (const float*)d_in[12];
    const float* W_val = (const float*)d_in[13];
    const float* b_val = (const float*)d_in[14];
    const float* W_p1  = (const float*)d_in[15];
    const float* b_p1  = (const float*)d_in[16];
    const float* W_p2  = (const float*)d_in[17];
    const float* b_p2  = (const float*)d_in[18];

    float* out   = (float*)d_out;
    float* preds = out;            // [16]
    float* w_out = out + NB;       // [317376]

    const int E = in_sizes[1];     // edges = B*N_GENES*DEG

    // workspace layout (floats)
    const size_t HTOT = (size_t)NNODES * DH;      // 20,312,064
    float* ws    = (float*)d_ws;
    float* h     = ws;                            // [NNODES*DH]
    float* acc   = ws + HTOT;                     // [NNODES*DH] (becomes h2 in-place)
    float* small = ws + 2 * HTOT;
    float* ssum  = small + 0;                     // [64]
    float* ssq   = small + 64;                    // [64]
    float* scale = small + 128;                   // [64]
    float* shift = small + 192;                   // [64]
    float* u     = small + 256;                   // [64]
    float* cbias = small + 320;                   // [1]
    float* S     = small + 384;                   // [16*64]
    float* wsum  = small + 384 + NB * DH;         // [16]
    v2f*   Bf    = (v2f*)(small + 384 + NB * DH + 16);  // [2048] float2 (16 KB)

    // K0: init + fold key->q + pre-swizzle W_gin into WMMA-B fragments
    k_init<<<1, 256, 0, stream>>>(W_key, b_key, W_q, W_gin, u, cbias, ssum, ssq,
                                  S, wsum, Bf);

    // K1: encoder
    {
        int total = NNODES * DH;
        k_encoder<<<(total + 255) / 256, 256, 0, stream>>>(x, W_enc, b_enc, gemb, h, acc);
    }
    // K2: scatter-add over edges (64 lanes per edge, coalesced f32 HW atomics)
    {
        int total = E * DH;
        k_scatter<<<(total + 255) / 256, 256, 0, stream>>>(h, esrc, edst, acc, total);
    }
    // K3: GIN GEMM with v_wmma_f32_16x16x4_f32, 8 waves/block, 1 tile/wave
    {
        int blocks = (NTILES + 7) / 8;
        k_gin_gemm<<<blocks, 256, 0, stream>>>(acc, Bf, ssum, ssq);
    }
    // K4: BN finalize
    k_bnfix<<<1, 64, 0, stream>>>(ssum, ssq, gamma, beta, scale, shift);

    // K5: fused BN+GELU+attention-weight+graph reduction
    {
        int blocks = (NTILES + 7) / 8;
        k_readout<<<blocks, 256, 0, stream>>>(acc, scale, shift, u, cbias, S, wsum, w_out);
    }
    // K6: tiny tail (value GEMM folded to 16 rows + predictor)
    k_final<<<1, 64, 0, stream>>>(S, wsum, W_val, b_val, W_p1, b_p1, W_p2, b_p2, preds);
}